// DMMRLoss_52621939310662
// MI455X (gfx1250) — compile-verified
//
#include <hip/hip_runtime.h>
#include <cmath>

typedef __attribute__((ext_vector_type(2))) float v2f;
typedef __attribute__((ext_vector_type(8))) float v8f;

#define PATCH 17
#define PVOL 4913          // 17^3
#define C1_STRIDE 520      // 512 padded to dodge LDS bank conflicts
#define NTHREADS 256

// ws int layout:
//  [0..2] minZ,minY,minX   [3..5] maxZ,maxY,maxX
//  [8..20] params: z0,y0,x0, sD,sH,sW, padD,padH,padW, nD,nH,nW, Npatch
// ws float layout: from index 32: val[maxP], keep[maxP]
#define WS_PARAM 8
#define WS_DATA 32

__global__ void k_init(int* wsI, int V) {
    if (blockIdx.x == 0 && threadIdx.x == 0) {
        wsI[0] = V; wsI[1] = V; wsI[2] = V;
        wsI[3] = -1; wsI[4] = -1; wsI[5] = -1;
    }
}

// bounding box of (target > 0): block-local shared-atomic reduce, then 6 global atomics
__global__ void k_bbox(const float* __restrict__ tgt, int V, int* wsI) {
    __shared__ int smn[3], smx[3];
    if (threadIdx.x == 0) { smn[0] = smn[1] = smn[2] = V; smx[0] = smx[1] = smx[2] = -1; }
    __syncthreads();
    size_t idx = (size_t)blockIdx.x * blockDim.x + threadIdx.x;
    size_t tot = (size_t)V * V * V;
    if (idx < tot) {
        if (tgt[idx] > 0.0f) {
            int x = (int)(idx % (size_t)V);
            size_t t = idx / (size_t)V;
            int y = (int)(t % (size_t)V);
            int z = (int)(t / (size_t)V);
            atomicMin(&smn[0], z); atomicMax(&smx[0], z);
            atomicMin(&smn[1], y); atomicMax(&smx[1], y);
            atomicMin(&smn[2], x); atomicMax(&smx[2], x);
        }
    }
    __syncthreads();
    if (threadIdx.x == 0) {
        atomicMin(&wsI[0], smn[0]); atomicMin(&wsI[1], smn[1]); atomicMin(&wsI[2], smn[2]);
        atomicMax(&wsI[3], smx[0]); atomicMax(&wsI[4], smx[1]); atomicMax(&wsI[5], smx[2]);
    }
}

__global__ void k_params(int* wsI) {
    if (blockIdx.x != 0 || threadIdx.x != 0) return;
    int z0 = wsI[0], y0 = wsI[1], x0 = wsI[2];
    int s[3] = { wsI[3] - z0, wsI[4] - y0, wsI[5] - x0 }; // slice [lo:hi] -> length hi-lo
    int* P = wsI + WS_PARAM;
    P[0] = z0; P[1] = y0; P[2] = x0;
    int nprod = 1;
    for (int d = 0; d < 3; ++d) {
        int sd = s[d] > 0 ? s[d] : 0;
        int r = sd % PATCH;
        int pad = (PATCH - r) % PATCH;
        P[3 + d] = sd;
        P[6 + d] = pad / 2;            // low-side pad
        int n = (sd + pad) / PATCH;
        P[9 + d] = n;
        nprod *= n;
    }
    P[12] = nprod;
}

// One workgroup (8 wave32) per patch: load+pad patch to LDS, zero-count,
// conv1 + conv2 via f32 WMMA implicit GEMM, fc1/fc2 on VALU, tanh, keep-mask.
// Static __shared__ (~118 KB) so the compiler validates the LDS footprint
// against the gfx1250 limit (320 KB/WG per CDNA5 ISA).
__global__ void __launch_bounds__(NTHREADS) k_patches(
    const float* __restrict__ src, const float* __restrict__ tgt,
    const float* __restrict__ w1,  const float* __restrict__ b1,
    const float* __restrict__ w2,  const float* __restrict__ b2,
    const float* __restrict__ fw1, const float* __restrict__ fb1,
    const float* __restrict__ fw2, const float* __restrict__ fb2,
    int V, const int* __restrict__ wsI, float* __restrict__ wsF, int maxP)
{
    __shared__ float sP [2 * PVOL];        // ch0 = fixed, ch1 = moving
    __shared__ float sW1[1728];            // 54*32, [k*32 + n]
    __shared__ float sC1[32 * C1_STRIDE];  // conv1 activations [ch][512 (+pad)]
    __shared__ float sC2[1728];            // 64*27, [ch*27 + m2] (== flatten order)
    __shared__ float sR [NTHREADS];        // reduction scratch
    __shared__ int sZ;

    const int tid = threadIdx.x;
    const int p   = blockIdx.x;
    const int* P  = wsI + WS_PARAM;
    const int Np  = P[12];
    float* valArr  = wsF + WS_DATA;
    float* keepArr = valArr + maxP;
    if (p >= Np) {                       // uniform across block; ws was poisoned -> write zeros
        if (tid == 0) { valArr[p] = 0.0f; keepArr[p] = 0.0f; }
        return;
    }
    const int z0 = P[0], y0 = P[1], x0 = P[2];
    const int sD = P[3], sH = P[4], sWd = P[5];
    const int pD = P[6], pH = P[7], pW = P[8];
    const int nH = P[10], nW = P[11];
    int pw = p % nW; int t0 = p / nW; int ph = t0 % nH; int pd = t0 / nH;

    if (tid == 0) sZ = 0;

    // conv1 weights -> LDS, k-major: sW1[k*32+n] = w1[n*54+k], k = ch*27 + dz*9+dy*3+dx
    for (int i = tid; i < 1728; i += NTHREADS) {
        int k = i >> 5, n = i & 31;
        sW1[i] = w1[n * 54 + k];
    }
    // gather padded patch (fixed & moving) -> LDS
    for (int i = tid; i < PVOL; i += NTHREADS) {
        int cx = i % PATCH; int t = i / PATCH; int cy = t % PATCH; int cz = t / PATCH;
        int Z = pd * PATCH + cz - pD;
        int Y = ph * PATCH + cy - pH;
        int X = pw * PATCH + cx - pW;
        float fv = 0.0f, mv2 = 0.0f;
        if (Z >= 0 && Z < sD && Y >= 0 && Y < sH && X >= 0 && X < sWd) {
            size_t g = ((size_t)(z0 + Z) * V + (size_t)(y0 + Y)) * V + (size_t)(x0 + X);
            fv  = tgt[g];
            mv2 = src[g];
        }
        sP[i] = fv; sP[PVOL + i] = mv2;
    }
    __syncthreads();

    // zero_frac on fixed patch (padding zeros included, as in reference)
    int cnt = 0;
    for (int i = tid; i < PVOL; i += NTHREADS) cnt += (sP[i] == 0.0f) ? 1 : 0;
    atomicAdd(&sZ, cnt);

    const int lane = tid & 31;
    const int wave = tid >> 5;
    const int l16  = lane & 15;
    const int lh   = lane >> 4;   // 0/1

    // ---- conv1 as implicit GEMM: (512 x 54) x (54 x 32), f32 WMMA 16x16x4 ----
    for (int i = 0; i < 8; ++i) {
        int t  = wave * 8 + i;          // 64 tiles: 32 M-tiles x 2 N-tiles
        int mt = t >> 1, nt = t & 1;
        int m  = mt * 16 + l16;         // A row this lane supplies
        int oz = m >> 6, oy = (m >> 3) & 7, ox = m & 7;
        int n  = nt * 16 + l16;         // B col this lane supplies
        v8f acc = {0.f,0.f,0.f,0.f,0.f,0.f,0.f,0.f};
        for (int kk = 0; kk < 14; ++kk) {   // K = 54 padded to 56
            v2f a, b;
            #pragma unroll
            for (int v = 0; v < 2; ++v) {
                int k = kk * 4 + v + 2 * lh;     // A: (M=l%16, K=v+2*(l/16)); B symmetric
                float av = 0.0f, bv = 0.0f;
                if (k < 54) {
                    int ch = (k >= 27) ? 1 : 0;
                    int tt = k - 27 * ch;
                    int dz = tt / 9; int r = tt - dz * 9;
                    int dy = r / 3;  int dx = r - dy * 3;
                    av = sP[ch * PVOL + ((2*oz + dz) * PATCH + (2*oy + dy)) * PATCH + (2*ox + dx)];
                    bv = sW1[k * 32 + n];
                }
                a[v] = av; b[v] = bv;
            }
            acc = __builtin_amdgcn_wmma_f32_16x16x4_f32(false, a, false, b,
                                                        (short)0, acc, false, false);
        }
        int ch = n;
        float bias = b1[ch];
        #pragma unroll
        for (int v = 0; v < 8; ++v) {        // D: (M = v + 8*(l/16), N = l%16)
            int mm = mt * 16 + v + 8 * lh;
            float r = acc[v] + bias;
            sC1[ch * C1_STRIDE + mm] = r > 0.0f ? r : 0.0f;
        }
    }
    __syncthreads();

    // ---- conv2 as implicit GEMM: (27(->32) x 864) x (864 x 64), one tile per wave ----
    {
        int mt = wave >> 2, nt = wave & 3;
        int m2 = mt * 16 + l16;              // 0..31, valid if < 27
        bool mvalid = m2 < 27;
        int oz = m2 / 9; int rm = m2 - oz * 9; int oy = rm / 3; int ox = rm - oy * 3;
        int n = nt * 16 + l16;               // output channel 0..63
        const float* w2r = w2 + (size_t)n * 864;   // w2[n][k], k = ch*27 + tap
        v8f acc = {0.f,0.f,0.f,0.f,0.f,0.f,0.f,0.f};
        for (int kk = 0; kk < 216; ++kk) {
            v2f a, b;
            #pragma unroll
            for (int v = 0; v < 2; ++v) {
                int k  = kk * 4 + v + 2 * lh;     // 0..863
                int ch = k / 27;
                int tt = k - ch * 27;
                int dz = tt / 9; int r2 = tt - dz * 9; int dy = r2 / 3; int dx = r2 - dy * 3;
                float av = 0.0f;
                if (mvalid)
                    av = sC1[ch * C1_STRIDE + ((2*oz + dz) * 8 + (2*oy + dy)) * 8 + (2*ox + dx)];
                a[v] = av;
                b[v] = w2r[k];
            }
            acc = __builtin_amdgcn_wmma_f32_16x16x4_f32(false, a, false, b,
                                                        (short)0, acc, false, false);
        }
        float bias = b2[n];
        #pragma unroll
        for (int v = 0; v < 8; ++v) {
            int mm = mt * 16 + v + 8 * lh;
            if (mm < 27) {
                float r = acc[v] + bias;
                sC2[n * 27 + mm] = r > 0.0f ? r : 0.0f;   // flatten order ch*27 + z*9+y*3+x
            }
        }
    }
    __syncthreads();

    // ---- fc1 (1728 -> 256) on VALU, one output per thread; fuse fc2 product ----
    {
        int j = tid;
        float s = fb1[j];
        const float* wr = fw1 + (size_t)j * 1728;
        for (int k = 0; k < 1728; k += 4) {
            float4 w4 = *(const float4*)(wr + k);
            s += sC2[k] * w4.x + sC2[k+1] * w4.y + sC2[k+2] * w4.z + sC2[k+3] * w4.w;
        }
        float y = s > 0.0f ? s : 0.0f;
        sR[j] = y * fw2[j];
    }
    __syncthreads();
    for (int off = NTHREADS / 2; off > 0; off >>= 1) {
        if (tid < off) sR[tid] += sR[tid + off];
        __syncthreads();
    }
    if (tid == 0) {
        float out  = tanhf(sR[0] + fb2[0]);
        float keep = (2 * sZ <= PVOL) ? 1.0f : 0.0f;   // zero_frac <= 0.5
        valArr[p]  = out * keep;
        keepArr[p] = keep;
    }
}

// Deterministic fixed-order reduction (no float atomics -> bit-stable replays)
__global__ void k_final(const float* __restrict__ wsF, int maxP, float* __restrict__ out) {
    if (blockIdx.x != 0 || threadIdx.x != 0) return;
    const float* valArr  = wsF + WS_DATA;
    const float* keepArr = valArr + maxP;
    float s = 0.0f, c = 0.0f;
    for (int i = 0; i < maxP; ++i) { s += valArr[i]; c += keepArr[i]; }
    out[0] = s / c;
}

extern "C" void kernel_launch(void* const* d_in, const int* in_sizes, int n_in,
                              void* d_out, int out_size, void* d_ws, size_t ws_size,
                              hipStream_t stream)
{
    const float* src = (const float*)d_in[0];
    const float* tgt = (const float*)d_in[1];
    const float* w1  = (const float*)d_in[2];
    const float* b1  = (const float*)d_in[3];
    const float* w2  = (const float*)d_in[4];
    const float* b2  = (const float*)d_in[5];
    const float* fw1 = (const float*)d_in[6];
    const float* fb1 = (const float*)d_in[7];
    const float* fw2 = (const float*)d_in[8];
    const float* fb2 = (const float*)d_in[9];

    int V = (int)llround(cbrt((double)in_sizes[0]));        // 222 for the given inputs
    int maxN = (V - 1 + PATCH - 1) / PATCH;                 // bbox slice length <= V-1
    int maxP = maxN * maxN * maxN;                          // 13^3 = 2197
    int* wsI = (int*)d_ws;
    float* wsF = (float*)d_ws;

    k_init<<<1, 1, 0, stream>>>(wsI, V);
    size_t nvox = (size_t)V * V * V;
    int nb = (int)((nvox + 255) / 256);
    k_bbox<<<nb, 256, 0, stream>>>(tgt, V, wsI);
    k_params<<<1, 1, 0, stream>>>(wsI);

    k_patches<<<maxP, NTHREADS, 0, stream>>>(src, tgt, w1, b1, w2, b2,
                                             fw1, fb1, fw2, fb2, V, wsI, wsF, maxP);
    k_final<<<1, 1, 0, stream>>>(wsF, maxP, (float*)d_out);
}